// HierarchicalAggregate_35639638622745
// MI455X (gfx1250) — compile-verified
//
#include <hip/hip_runtime.h>

typedef float v2f __attribute__((ext_vector_type(2)));
typedef float v8f __attribute__((ext_vector_type(8)));

constexpr int EMB = 128;
constexpr int LDS_STRIDE = 132;   // 128 + 4 pad -> conflict-free ds_load_b64
constexpr int A_ROWS = 128;       // M rows per block (4 waves x 32)
constexpr int B_ROWS = 80;        // N cols per block (5 tiles x 16)
constexpr size_t LDS_BYTES = (size_t)(A_ROWS + B_ROWS) * LDS_STRIDE * sizeof(float);

// ---------------- zero the aggregation buffer (ws is poisoned) ----------------
__global__ __launch_bounds__(256) void ha_zero(float* __restrict__ p, int n) {
    int i = blockIdx.x * blockDim.x + threadIdx.x;
    if (i < n) p[i] = 0.0f;
}

// ---------------- SpMM scatter: one wave (32 lanes) per nonzero ----------------
__global__ __launch_bounds__(256) void ha_scatter(
    const float* __restrict__ W, const float* __restrict__ vals,
    const int* __restrict__ rows, const int* __restrict__ cols,
    float* __restrict__ agg, int nnz) {
    int gid  = blockIdx.x * blockDim.x + threadIdx.x;
    int e    = gid >> 5;
    int lane = gid & 31;
    if (e >= nnz) return;
    int   c = cols[e];
    int   r = rows[e];
    float v = vals[e];
    const float4 wv = *(const float4*)(W + (size_t)c * EMB + lane * 4);
    float* dst = agg + (size_t)r * EMB + lane * 4;
    unsafeAtomicAdd(dst + 0, v * wv.x);   // global_atomic_add_f32 (L2-resident agg)
    unsafeAtomicAdd(dst + 1, v * wv.y);
    unsafeAtomicAdd(dst + 2, v * wv.z);
    unsafeAtomicAdd(dst + 3, v * wv.w);
}

// ---------------- CDNA5 async-copy helpers ----------------
__device__ __forceinline__ uint32_t lds_off(const void* p) {
    return (uint32_t)(uintptr_t)p;   // low 32 bits of generic LDS ptr = LDS offset
}
__device__ __forceinline__ void async_copy_b128(uint32_t lds_byte, const float* gaddr) {
    asm volatile("global_load_async_to_lds_b128 %0, %1, off"
                 :: "v"(lds_byte), "v"(gaddr) : "memory");
}
__device__ __forceinline__ void wait_async0() {
    asm volatile("s_wait_asynccnt 0x0" ::: "memory");
}

__device__ __forceinline__ v8f wmma_f32(v2f a, v2f b, v8f c) {
    return __builtin_amdgcn_wmma_f32_16x16x4_f32(false, a, false, b, (short)0, c,
                                                 false, false);
}

// ---------------- GEMM: out[B,N] = x[B,128] @ agg[N,128]^T + b[N] ----------------
// Block = 4 waves (128 thr). Wave tile = 32(M) x 80(N): MT=2 x NT=5, 10 accumulators.
// A panel (128x128) + B panel (80x128) async-staged to padded LDS once per block.
// Software-pipelined k-loop: frags for k+1 load before the 10 WMMAs of step k.
__global__ __launch_bounds__(128) void ha_gemm_wmma(
    const float* __restrict__ X, const float* __restrict__ agg,
    const float* __restrict__ bias, float* __restrict__ out, int Ncols) {
    extern __shared__ float smem[];
    float* As = smem;                       // [128][132]
    float* Bs = smem + A_ROWS * LDS_STRIDE; // [80][132]

    const int lane = threadIdx.x & 31;
    const int wave = threadIdx.x >> 5;      // 0..3
    const int lo   = lane & 15;
    const int hi   = lane >> 4;

    const int m_block = blockIdx.y * A_ROWS;
    const int n_base  = blockIdx.x * B_ROWS;

    // ---- stage panels: one async b128 instruction moves one 512B row ----
    for (int r = wave * 32; r < wave * 32 + 32; ++r) {        // A rows
        async_copy_b128(lds_off(&As[r * LDS_STRIDE + lane * 4]),
                        X + (size_t)(m_block + r) * EMB + lane * 4);
    }
    for (int r = wave * 20; r < wave * 20 + 20; ++r) {        // B rows
        async_copy_b128(lds_off(&Bs[r * LDS_STRIDE + lane * 4]),
                        agg + (size_t)(n_base + r) * EMB + lane * 4);
    }
    wait_async0();
    __syncthreads();

    // ---- compute: MT=2 x NT=5 ----
    v8f acc0[5], acc1[5];
#pragma unroll
    for (int t = 0; t < 5; ++t) {
        acc0[t] = (v8f){0, 0, 0, 0, 0, 0, 0, 0};
        acc1[t] = (v8f){0, 0, 0, 0, 0, 0, 0, 0};
    }

    const float* arow0 = &As[(wave * 32 + lo) * LDS_STRIDE + 2 * hi];
    const float* arow1 = arow0 + 16 * LDS_STRIDE;
    const float* brow[5];
#pragma unroll
    for (int t = 0; t < 5; ++t)
        brow[t] = &Bs[(16 * t + lo) * LDS_STRIDE + 2 * hi];

    // prologue: fragments for k-step 0
    v2f a0 = *(const v2f*)(arow0);
    v2f a1 = *(const v2f*)(arow1);
    v2f bf[5];
#pragma unroll
    for (int t = 0; t < 5; ++t) bf[t] = *(const v2f*)(brow[t]);

#pragma unroll 4
    for (int kk = 0; kk < EMB / 4 - 1; ++kk) {
        // prefetch k+1 fragments (7 ds loads) ahead of the 10 WMMAs
        v2f na0 = *(const v2f*)(arow0 + 4 * (kk + 1));
        v2f na1 = *(const v2f*)(arow1 + 4 * (kk + 1));
        v2f nb[5];
#pragma unroll
        for (int t = 0; t < 5; ++t) nb[t] = *(const v2f*)(brow[t] + 4 * (kk + 1));
#pragma unroll
        for (int t = 0; t < 5; ++t) {
            acc0[t] = wmma_f32(a0, bf[t], acc0[t]);
            acc1[t] = wmma_f32(a1, bf[t], acc1[t]);
        }
        a0 = na0; a1 = na1;
#pragma unroll
        for (int t = 0; t < 5; ++t) bf[t] = nb[t];
    }
    // epilogue k-step
#pragma unroll
    for (int t = 0; t < 5; ++t) {
        acc0[t] = wmma_f32(a0, bf[t], acc0[t]);
        acc1[t] = wmma_f32(a1, bf[t], acc1[t]);
    }

    // ---- bias + store ----
#pragma unroll
    for (int t = 0; t < 5; ++t) {
        const int col = n_base + 16 * t + lo;
        const float bv = bias[col];
        const int row0 = m_block + wave * 32 + 8 * hi;
#pragma unroll
        for (int r = 0; r < 8; ++r)
            out[(size_t)(row0 + r) * Ncols + col] = acc0[t][r] + bv;
#pragma unroll
        for (int r = 0; r < 8; ++r)
            out[(size_t)(row0 + 16 + r) * Ncols + col] = acc1[t][r] + bv;
    }
}

extern "C" void kernel_launch(void* const* d_in, const int* in_sizes, int n_in,
                              void* d_out, int out_size, void* d_ws, size_t ws_size,
                              hipStream_t stream) {
    const float* x    = (const float*)d_in[0];
    const float* W    = (const float*)d_in[1];
    const float* b    = (const float*)d_in[2];
    const float* vals = (const float*)d_in[3];
    const int*   rows = (const int*)d_in[4];
    const int*   cols = (const int*)d_in[5];
    float*       out  = (float*)d_out;

    const int nconc = in_sizes[2];       // 50000
    const int nnz   = in_sizes[3];       // 800000
    const int batch = in_sizes[0] / EMB; // 4096

    float* agg = (float*)d_ws;           // 25.6 MB scratch
    const int aggN = nconc * EMB;

    ha_zero<<<(aggN + 255) / 256, 256, 0, stream>>>(agg, aggN);

    const long sthreads = (long)nnz * 32;
    ha_scatter<<<(int)((sthreads + 255) / 256), 256, 0, stream>>>(
        W, vals, rows, cols, agg, nnz);

    dim3 grid(nconc / B_ROWS, batch / A_ROWS);   // (625, 32)
    ha_gemm_wmma<<<grid, 128, LDS_BYTES, stream>>>(x, agg, b, out, nconc);
}